// Model_62148176773314
// MI455X (gfx1250) — compile-verified
//
#include <hip/hip_runtime.h>
#include <math.h>

typedef float v2f __attribute__((ext_vector_type(2)));
typedef float v8f __attribute__((ext_vector_type(8)));

#define N1 32
#define N2 64
#define PP 196
#define DD 128
#define PHH 14
#define PWW 14
#define HH 224
#define WW 224

// workspace layout (float offsets)
#define OFF_INVF   0
#define OFF_INVNF  (OFF_INVF + N1*PP)            // 6272
#define OFF_SP     (OFF_INVNF + N2*PP)           // 18816
#define OFF_SBITS  (OFF_SP + N1*N2*PP)           // 420224
#define OFF_SPMEAN (OFF_SBITS + N1*N2)           // 422272

// ---------------- kernel 0: zero score bits ----------------
__global__ void k_init(unsigned int* __restrict__ sbits) {
    int i = blockIdx.x * blockDim.x + threadIdx.x;
    if (i < N1 * N2) sbits[i] = 0u;
}

// ---------------- kernel 1: inverse L2 norms (one wave per row) ----------------
__global__ void k_norms(const float* __restrict__ feat,
                        const float* __restrict__ nfeat,
                        float* __restrict__ ws) {
    int gw   = (blockIdx.x * blockDim.x + threadIdx.x) >> 5;
    int lane = threadIdx.x & 31;
    const int nrows = N1 * PP + N2 * PP;
    if (gw >= nrows) return;
    const float* src = (gw < N1 * PP) ? (feat + (size_t)gw * DD)
                                      : (nfeat + (size_t)(gw - N1 * PP) * DD);
    float s = 0.f;
#pragma unroll
    for (int i = 0; i < 4; ++i) {
        float v = src[lane + 32 * i];
        s += v * v;
    }
#pragma unroll
    for (int off = 16; off >= 1; off >>= 1) s += __shfl_xor(s, off, 32);
    float inv = rsqrtf(s);
    inv = inv * (1.5f - 0.5f * s * inv * inv);   // Newton refinement
    if (lane == 0) ws[gw] = inv;                 // invf then invnf, contiguous
}

// ---------------- kernel 2: fused normalized GEMM + max-over-q ----------------
__global__ __launch_bounds__(256) void k_gemm(
        const float* __restrict__ feat, const float* __restrict__ nfeat,
        const float* __restrict__ bgmask, float* __restrict__ ws) {
    // B tile of nf, pair-major: logical (k,q) -> sB[(k>>1)*32 + q*2 + (k&1)]
    __shared__ float sB[16 * DD];                // 8 KB

    const int a    = blockIdx.y;
    const int b    = blockIdx.x;
    const int tid  = threadIdx.x;
    const int wave = tid >> 5;
    const int lane = tid & 31;
    const int h    = lane >> 4;                  // half-wave select (K group)
    const int n    = lane & 15;                  // N / M-in-tile index

    const float* invf  = ws + OFF_INVF;
    const float* invnf = ws + OFF_INVNF;
    float*       sp    = ws + OFF_SP;
    unsigned int* sbits = (unsigned int*)(ws + OFF_SBITS);

    const float* fa  = feat  + (size_t)a * PP * DD;
    const float* nfb = nfeat + (size_t)b * PP * DD;
    const v2f*   sB2 = (const v2f*)sB;

    for (int pass = 0; pass < 2; ++pass) {
        const int  pt    = wave + pass * 8;      // p-tile id, 13 tiles total
        const bool valid = (pt < 13);
        const int  ptc   = valid ? pt : 0;

        // A fragment: row M = n of tile, K striped per WMMA layout
        const int    prow = min(ptc * 16 + n, PP - 1);   // clamp: dup rows discarded at store
        const float  iva  = invf[a * PP + prow];
        const float* arow = fa + (size_t)prow * DD + 2 * h;
        v2f Af[32];
#pragma unroll
        for (int s = 0; s < 32; ++s) {
            v2f v = *(const v2f*)(arow + 4 * s);
            Af[s] = v * iva;
        }

        float rmax[8];
#pragma unroll
        for (int r = 0; r < 8; ++r) rmax[r] = -3.0e38f;

        for (int qt = 0; qt < 13; ++qt) {
            __syncthreads();
            // cooperative stage of B tile (coalesced global, conflict-free LDS)
#pragma unroll
            for (int i = 0; i < 8; ++i) {
                int e  = i * 256 + tid;          // 0..2047
                int q  = e >> 7;                 // 0..15
                int k  = e & 127;
                int qg = min(qt * 16 + q, PP - 1);   // clamp: dup columns can't change a max
                float v = nfb[(size_t)qg * DD + k] * invnf[b * PP + qg];
                sB[(k >> 1) * 32 + q * 2 + (k & 1)] = v;
            }
            __syncthreads();

            // Two independent accumulation chains (even/odd k-steps) to double
            // per-wave WMMA ILP; exact since K-accumulation is a plain sum.
            v8f c0 = {0.f, 0.f, 0.f, 0.f, 0.f, 0.f, 0.f, 0.f};
            v8f c1 = {0.f, 0.f, 0.f, 0.f, 0.f, 0.f, 0.f, 0.f};
#pragma unroll
            for (int s = 0; s < 16; ++s) {
                v2f bF0 = sB2[(2 * (2 * s)     + h) * 16 + n];
                v2f bF1 = sB2[(2 * (2 * s + 1) + h) * 16 + n];
                c0 = __builtin_amdgcn_wmma_f32_16x16x4_f32(
                        false, Af[2 * s],     false, bF0, (short)0, c0, false, false);
                c1 = __builtin_amdgcn_wmma_f32_16x16x4_f32(
                        false, Af[2 * s + 1], false, bF1, (short)0, c1, false, false);
            }
#pragma unroll
            for (int r = 0; r < 8; ++r) rmax[r] = fmaxf(rmax[r], c0[r] + c1[r]);
        }

        // max over the 16 N-lanes of each half-wave (C layout: halves hold M and M+8)
#pragma unroll
        for (int off = 1; off < 16; off <<= 1) {
#pragma unroll
            for (int r = 0; r < 8; ++r)
                rmax[r] = fmaxf(rmax[r], __shfl_xor(rmax[r], off, 32));
        }

        if (valid && n == 0) {
#pragma unroll
            for (int r = 0; r < 8; ++r) {
                int p = pt * 16 + r + 8 * h;
                if (p < PP) {
                    float sv = rmax[r];
                    float d  = sqrtf(fmaxf(2.f - 2.f * sv, 0.f)) * 0.5f;
                    float v  = d * bgmask[a * PP + p];
                    sp[((size_t)(a * N2 + b)) * PP + p] = v;
                    atomicMax(&sbits[a * N2 + b], __float_as_uint(v)); // v >= 0
                }
            }
        }
    }
}

// ---------------- kernel 3: mean over b ----------------
__global__ void k_meanb(float* __restrict__ ws) {
    int i = blockIdx.x * blockDim.x + threadIdx.x;
    if (i >= N1 * PP) return;
    int a = i / PP, p = i % PP;
    const float* sp = ws + OFF_SP;
    float s = 0.f;
    for (int b = 0; b < N2; ++b) s += sp[((size_t)(a * N2 + b)) * PP + p];
    ws[OFF_SPMEAN + i] = s * (1.0f / N2);
}

// ---------------- kernel 4: bilinear 14x14 -> 224x224 (align_corners=False) --
__global__ void k_bilinear(const float* __restrict__ ws, float* __restrict__ out) {
    int idx = blockIdx.x * blockDim.x + threadIdx.x;
    if (idx >= N1 * HH * WW) return;
    int x = idx % WW;
    int y = (idx / WW) % HH;
    int a = idx / (HH * WW);
    const float* m = ws + OFF_SPMEAN + a * PP;
    float sy = fmaxf((y + 0.5f) * ((float)PHH / (float)HH) - 0.5f, 0.f);
    float sx = fmaxf((x + 0.5f) * ((float)PWW / (float)WW) - 0.5f, 0.f);
    int y0 = min((int)sy, PHH - 1); int y1 = min(y0 + 1, PHH - 1); float wy = sy - (float)y0;
    int x0 = min((int)sx, PWW - 1); int x1 = min(x0 + 1, PWW - 1); float wx = sx - (float)x0;
    float v00 = m[y0 * PWW + x0], v01 = m[y0 * PWW + x1];
    float v10 = m[y1 * PWW + x0], v11 = m[y1 * PWW + x1];
    float v = (1.f - wy) * ((1.f - wx) * v00 + wx * v01)
            +        wy  * ((1.f - wx) * v10 + wx * v11);
    out[N1 + idx] = v;
}

// ---------------- kernel 5: scores.mean over b ----------------
__global__ void k_scores(const float* __restrict__ ws, float* __restrict__ out) {
    int a = threadIdx.x;
    if (a >= N1) return;
    const unsigned int* sbits = (const unsigned int*)(ws + OFF_SBITS);
    float s = 0.f;
    for (int b = 0; b < N2; ++b) s += __uint_as_float(sbits[a * N2 + b]);
    out[a] = s * (1.0f / N2);
}

extern "C" void kernel_launch(void* const* d_in, const int* in_sizes, int n_in,
                              void* d_out, int out_size, void* d_ws, size_t ws_size,
                              hipStream_t stream) {
    const float* feat   = (const float*)d_in[0];
    const float* nfeat  = (const float*)d_in[1];
    const float* bgmask = (const float*)d_in[2];
    float* ws  = (float*)d_ws;
    float* out = (float*)d_out;

    k_init<<<(N1 * N2 + 255) / 256, 256, 0, stream>>>(
        (unsigned int*)(ws + OFF_SBITS));

    const int nrows = N1 * PP + N2 * PP;
    k_norms<<<(nrows * 32 + 255) / 256, 256, 0, stream>>>(feat, nfeat, ws);

    dim3 grid(N2, N1);
    k_gemm<<<grid, 256, 0, stream>>>(feat, nfeat, bgmask, ws);

    k_meanb<<<(N1 * PP + 255) / 256, 256, 0, stream>>>(ws);
    k_bilinear<<<(N1 * HH * WW + 255) / 256, 256, 0, stream>>>(ws, out);
    k_scores<<<1, 32, 0, stream>>>(ws, out);
}